// VectorQuantizer_23854248362089
// MI455X (gfx1250) — compile-verified
//
#include <hip/hip_runtime.h>

typedef __attribute__((ext_vector_type(2))) float v2f;
typedef __attribute__((ext_vector_type(8))) float v8f;

#define N_E   1024
#define E_DIM 256
#define NBATCH 32
#define HW    1024              // 32*32
#define N_TOT (NBATCH * HW)     // 32768
#define BETA  0.25f

// ws layout (in floats)
#define WS_EMBT  0                    // 256*1024 floats
#define WS_ENORM (E_DIM * N_E)        // 1024 floats
#define WS_IDX   (WS_ENORM + N_E)     // 32768 ints

// d_out layout (in floats)
#define OUT_LOSS (NBATCH * E_DIM * HW)  // 8388608
#define OUT_IDX  (OUT_LOSS + 1)

// ---------------------------------------------------------------------------
// Prep: embT[c][k] = emb[k][c]; enorm[k] = ||emb[k]||^2; zero the loss scalar.
// ---------------------------------------------------------------------------
__global__ void __launch_bounds__(E_DIM)
vq_prep(const float* __restrict__ emb, float* __restrict__ embT,
        float* __restrict__ enorm, float* __restrict__ loss_out) {
    const int k = blockIdx.x;    // code
    const int c = threadIdx.x;   // channel
    float v = emb[k * E_DIM + c];
    embT[c * N_E + k] = v;
    float s = v * v;
    #pragma unroll
    for (int m = 16; m >= 1; m >>= 1) s += __shfl_xor(s, m, 32);
    __shared__ float red[8];
    if ((threadIdx.x & 31) == 0) red[threadIdx.x >> 5] = s;
    __syncthreads();
    if (threadIdx.x == 0) {
        float t = 0.f;
        #pragma unroll
        for (int i = 0; i < 8; ++i) t += red[i];
        enorm[k] = t;
        if (blockIdx.x == 0) *loss_out = 0.f;
    }
}

// ---------------------------------------------------------------------------
// Argmin: 256-row tile per block vs all 1024 codes via f32 WMMA.
//   LDS holds channel-PAIR interleaved tiles so every WMMA fragment (A and B)
//   is one aligned ds_load_b64 into an adjacent VGPR pair:
//     Ail[cp*256 + j] = { z[2cp][j],   z[2cp+1][j] }      (256 KB)
//     Bil[cp*64  + k] = { eT[2cp][k],  eT[2cp+1][k] }     ( 64 KB chunk)
//   Each wave: 32 rows (2 stripes) x 64 codes (4 tiles) -> 8 WMMA / 6 DS-b64
//   per K-step. Relative distance d(n,k) = enorm[k] - 2 * dot(z_n, e_k).
// ---------------------------------------------------------------------------
__global__ void __launch_bounds__(256)
vq_argmin(const float* __restrict__ z, const float* __restrict__ embT,
          const float* __restrict__ enorm, int* __restrict__ idx_i,
          float* __restrict__ idx_f) {
    extern __shared__ float smem[];                 // 320 KB total
    float2* Ail = (float2*)smem;                    // 32768 float2 = 256 KB
    float2* Bil = (float2*)(smem + 65536);          //  8192 float2 =  64 KB

    const int bid = blockIdx.x;                     // 128 blocks, 256 rows each
    const int t   = threadIdx.x;
    const int b   = bid >> 2;
    const int hw0 = (bid & 3) << 8;
    const float* zbase = z + ((size_t)b * E_DIM) * HW + hw0;

    // ---- Stage A interleaved (coalesced float4 reads, b64 LDS writes) ------
    for (int i = t; i < 8192; i += 256) {           // 128 ch-pairs x 64 float4
        const int cp = i >> 6;
        const int j4 = i & 63;
        const float4 fa = ((const float4*)(zbase + (2 * cp)     * HW))[j4];
        const float4 fb = ((const float4*)(zbase + (2 * cp + 1) * HW))[j4];
        float2* d = Ail + cp * 256 + j4 * 4;
        d[0] = make_float2(fa.x, fb.x);
        d[1] = make_float2(fa.y, fb.y);
        d[2] = make_float2(fa.z, fb.z);
        d[3] = make_float2(fa.w, fb.w);
    }

    const int lane = t & 31;
    const int wv   = t >> 5;        // wave id 0..7 -> 32-row stripe pair
    const int j0   = wv << 5;
    const int half = lane >> 4;
    const int l16  = lane & 15;

    float minv[16];
    int   mini[16];
    #pragma unroll
    for (int v = 0; v < 16; ++v) { minv[v] = 3.4e38f; mini[v] = 0; }

    // per-lane fragment bases (float2 units); per K-step: A +512, B +128
    const float2* Abase0 = Ail + half * 256 + j0 + l16;
    const float2* Abase1 = Abase0 + 16;
    const float2* Bbase  = Bil + half * 64 + l16;

    for (int g = 0; g < 16; ++g) {                  // 16 groups of 64 codes
        const int k0 = g << 6;
        __syncthreads();        // prior reads done (g>0); A staging done (g==0)

        // ---- Stage B chunk interleaved ----
        for (int i = t; i < 2048; i += 256) {       // 128 ch-pairs x 16 float4
            const int cp = i >> 4;
            const int k4 = i & 15;
            const float4 fa = ((const float4*)(embT + (size_t)(2 * cp)     * N_E + k0))[k4];
            const float4 fb = ((const float4*)(embT + (size_t)(2 * cp + 1) * N_E + k0))[k4];
            float2* d = Bil + cp * 64 + k4 * 4;
            d[0] = make_float2(fa.x, fb.x);
            d[1] = make_float2(fa.y, fb.y);
            d[2] = make_float2(fa.z, fb.z);
            d[3] = make_float2(fa.w, fb.w);
        }
        __syncthreads();

        v8f acc[2][4];
        #pragma unroll
        for (int s = 0; s < 2; ++s)
            #pragma unroll
            for (int tile = 0; tile < 4; ++tile) acc[s][tile] = {};

        // ---- K = 256: 4 chunks of 16 steps (all DS offsets are immediates) -
        #pragma unroll
        for (int qk = 0; qk < 4; ++qk) {
            const float2* Ap0 = Abase0 + qk * (16 * 512);
            const float2* Ap1 = Abase1 + qk * (16 * 512);
            const float2* Bp  = Bbase  + qk * (16 * 128);
            #pragma unroll
            for (int k2 = 0; k2 < 16; ++k2) {
                const v2f a0 = *(const v2f*)(Ap0 + k2 * 512);   // ds_load_b64
                const v2f a1 = *(const v2f*)(Ap1 + k2 * 512);
                #pragma unroll
                for (int tile = 0; tile < 4; ++tile) {
                    const v2f bb = *(const v2f*)(Bp + k2 * 128 + tile * 16);
                    acc[0][tile] = __builtin_amdgcn_wmma_f32_16x16x4_f32(
                        false, a0, false, bb, (short)0, acc[0][tile], false, false);
                    acc[1][tile] = __builtin_amdgcn_wmma_f32_16x16x4_f32(
                        false, a1, false, bb, (short)0, acc[1][tile], false, false);
                }
            }
        }

        // ---- Fold into running min/argmin (row = v + 8*half [+16*s], N = l16)
        #pragma unroll
        for (int tile = 0; tile < 4; ++tile) {
            const int kg = k0 + tile * 16 + l16;
            const float en = enorm[kg];
            #pragma unroll
            for (int s = 0; s < 2; ++s) {
                #pragma unroll
                for (int v = 0; v < 8; ++v) {
                    const float d = en - 2.0f * acc[s][tile][v];
                    const int m = s * 8 + v;
                    if (d < minv[m]) { minv[m] = d; mini[m] = kg; }
                }
            }
        }
    }

    // ---- Butterfly reduce across the 16 lanes of each half ------------------
    #pragma unroll
    for (int v = 0; v < 16; ++v) {
        #pragma unroll
        for (int m = 1; m <= 8; m <<= 1) {
            const float ov = __shfl_xor(minv[v], m, 32);
            const int   oi = __shfl_xor(mini[v], m, 32);
            if (ov < minv[v] || (ov == minv[v] && oi < mini[v])) { minv[v] = ov; mini[v] = oi; }
        }
    }
    if (l16 == 0) {
        #pragma unroll
        for (int s = 0; s < 2; ++s)
            #pragma unroll
            for (int v = 0; v < 8; ++v) {
                const int n = bid * 256 + j0 + s * 16 + half * 8 + v;
                idx_i[n] = mini[s * 8 + v];
                idx_f[n] = (float)mini[s * 8 + v];
            }
    }
}

// ---------------------------------------------------------------------------
// Output + loss: z_q[b][c][hw] = embT[c][idx[b*1024+hw]]; accumulate BETA*mse.
// ---------------------------------------------------------------------------
__global__ void __launch_bounds__(256)
vq_out(const float* __restrict__ z, const float* __restrict__ embT,
       const int* __restrict__ idx_i, float* __restrict__ zq,
       float* __restrict__ loss) {
    const int blk = blockIdx.x;          // 8192 = 32 * 256
    const int b = blk >> 8;
    const int c = blk & 255;
    const size_t base = ((size_t)b * E_DIM + c) * HW;
    const float* zr   = z + base;
    float*       outr = zq + base;
    const int*   ir   = idx_i + b * HW;
    const float* er   = embT + (size_t)c * N_E;

    float part = 0.f;
    #pragma unroll
    for (int r = 0; r < 4; ++r) {
        const int hw = threadIdx.x + (r << 8);
        const float q = er[ir[hw]];
        outr[hw] = q;
        const float d = q - zr[hw];
        part += d * d;
    }
    #pragma unroll
    for (int m = 16; m >= 1; m >>= 1) part += __shfl_xor(part, m, 32);
    __shared__ float red[8];
    if ((threadIdx.x & 31) == 0) red[threadIdx.x >> 5] = part;
    __syncthreads();
    if (threadIdx.x == 0) {
        float s = 0.f;
        #pragma unroll
        for (int i = 0; i < 8; ++i) s += red[i];
        atomicAdd(loss, s * (BETA / (float)(N_TOT * E_DIM)));
    }
}

// ---------------------------------------------------------------------------
extern "C" void kernel_launch(void* const* d_in, const int* in_sizes, int n_in,
                              void* d_out, int out_size, void* d_ws, size_t ws_size,
                              hipStream_t stream) {
    (void)in_sizes; (void)n_in; (void)out_size; (void)ws_size;
    const float* z   = (const float*)d_in[0];
    const float* emb = (const float*)d_in[1];

    float* ws    = (float*)d_ws;
    float* embT  = ws + WS_EMBT;
    float* enorm = ws + WS_ENORM;
    int*   idx_i = (int*)(ws + WS_IDX);

    float* out   = (float*)d_out;
    float* zq    = out;
    float* loss  = out + OUT_LOSS;
    float* idx_f = out + OUT_IDX;

    vq_prep<<<N_E, E_DIM, 0, stream>>>(emb, embT, enorm, loss);

    const size_t smem = (size_t)(E_DIM * 256 + E_DIM * 64) * sizeof(float); // 320KB
    hipFuncSetAttribute(reinterpret_cast<const void*>(vq_argmin),
                        hipFuncAttributeMaxDynamicSharedMemorySize, (int)smem);
    vq_argmin<<<N_TOT / 256, 256, smem, stream>>>(z, embT, enorm, idx_i, idx_f);

    vq_out<<<NBATCH * E_DIM, 256, 0, stream>>>(z, embT, idx_i, zq, loss);
}